// PamDelayModel_36790689858174
// MI455X (gfx1250) — compile-verified
//
#include <hip/hip_runtime.h>
#include <cstdint>

#define RING_L 22
#define CHUNK  32
#define BLOCK  256
#define DT_F   0.005f

// ---- CDNA5 async global->LDS helpers (ASYNCcnt-tracked) -------------------
// Low 32 bits of a flat LDS pointer are the LDS byte offset (aperture holds
// the base in the upper 32 bits) -> exactly what the VDST operand of
// global_load_async_to_lds_* wants.
__device__ __forceinline__ uint32_t lds_byte_off(const void* p) {
  return (uint32_t)(uintptr_t)p;
}

// GVS form: mem = SGPR64 base + signed VGPR32 byte offset. One B32 per lane,
// 128B per wave per issue, tracked on ASYNCcnt (completes in order).
__device__ __forceinline__ void async_g2l_b32(uint32_t lds_off, const float* sbase,
                                              int voff_bytes) {
  asm volatile("global_load_async_to_lds_b32 %0, %1, %2"
               :
               : "v"(lds_off), "v"(voff_bytes), "s"(sbase)
               : "memory");
}

template <int N>
__device__ __forceinline__ void wait_asynccnt_le() {
  asm volatile("s_wait_asynccnt %0" ::"n"(N) : "memory");
}

// ---------------------------------------------------------------------------
__global__ __launch_bounds__(BLOCK) void pam_delay_kernel(
    const float* __restrict__ in,   // [T, N] flattened (N = E*C)
    const float* __restrict__ dax, const float* __restrict__ dvl,
    const float* __restrict__ tax, const float* __restrict__ tvl,
    float* __restrict__ out,        // [T, N]
    int T, int N) {
  extern __shared__ float sm[];
  float*  stage = sm;                                  // [2][CHUNK][BLOCK]
  float*  ring  = sm + 2 * CHUNK * BLOCK;              // [RING_L][BLOCK]
  float2* coef  = (float2*)(sm + 2 * CHUNK * BLOCK + RING_L * BLOCK); // [2][8]

  const int tid = threadIdx.x;
  int g = blockIdx.x * BLOCK + tid;
  const bool active = (g < N);
  if (g >= N) g = N - 1;  // duplicated loads harmless; stores guarded

  // Zero this thread's ring column (reference: buf0 = zeros). Own column only.
#pragma unroll
  for (int i = 0; i < RING_L; ++i) ring[i * BLOCK + tid] = 0.0f;

  // Per-segment affine coefficients: y = a + b*xc
  //   b = (f1-f0)/(x1-x0+1e-12),  a = f0 - b*x0   (== f0 + t*(f1-f0))
  if (tid < 5) {
    {
      float x0 = dax[tid], x1 = dax[tid + 1];
      float f0 = dvl[tid], f1 = dvl[tid + 1];
      float b = (f1 - f0) / (x1 - x0 + 1e-12f);
      coef[tid] = make_float2(f0 - b * x0, b);
    }
    {
      float x0 = tax[tid], x1 = tax[tid + 1];
      float f0 = tvl[tid], f1 = tvl[tid + 1];
      float b = (f1 - f0) / (x1 - x0 + 1e-12f);
      coef[8 + tid] = make_float2(f0 - b * x0, b);
    }
  }

  // Axis thresholds (uniform, cache-resident).
  const float d_lo = dax[0], d_hi = dax[5];
  const float d1 = dax[1], d2 = dax[2], d3 = dax[3], d4 = dax[4];
  const float t_lo = tax[0], t_hi = tax[5];
  const float ta1 = tax[1], ta2 = tax[2], ta3 = tax[3], ta4 = tax[4];

  __syncthreads();  // coef visible to all waves

  const uint32_t stage_b0 = lds_byte_off(stage) + (uint32_t)tid * 4u;
  const int row_b = N * 4;  // global byte stride between timesteps

  // Issue one chunk: per-lane async B32 column copies (coalesced 128B/row/wave).
  auto issue_chunk = [&](int c, int buf) {
    const int t0 = c * CHUNK;
    const uint32_t lb = stage_b0 + (uint32_t)buf * (CHUNK * BLOCK * 4);
    const int vb = (t0 * N + g) * 4;  // base byte offset, scalar*row + lane
#pragma unroll
    for (int s = 0; s < CHUNK; ++s) {
      if (t0 + s < T)
        async_g2l_b32(lb + (uint32_t)s * (BLOCK * 4), in, vb + s * row_b);
    }
  };

  const int nch = (T + CHUNK - 1) / CHUNK;
  issue_chunk(0, 0);

  float cur = 0.0f;
  int ptr = 0;
  const float invDT = 1.0f / DT_F;

  for (int c = 0; c < nch; ++c) {
    const int buf = c & 1;
    if (c + 1 < nch) {
      issue_chunk(c + 1, buf ^ 1);  // overlap next chunk's loads with compute
      wait_asynccnt_le<CHUNK>();    // in-order: current chunk's loads are done
    } else {
      wait_asynccnt_le<0>();
    }

    const float* st = stage + buf * (CHUNK * BLOCK);
    const int t0 = c * CHUNK;
#pragma unroll 8
    for (int s = 0; s < CHUNK; ++s) {
      const int t = t0 + s;
      if (t >= T) break;
      const float x = st[s * BLOCK + tid];

      // dead = interp_lut(x, dax, dvl)   (searchsorted left, clip [1,5])
      float xc = fminf(fmaxf(x, d_lo), d_hi);
      int di = (xc > d1) + (xc > d2) + (xc > d3) + (xc > d4);
      float2 cd = coef[di];
      float dead = fmaf(cd.y, xc, cd.x);

      // tau = interp_lut(x, tax, tvl)
      float yc = fminf(fmaxf(x, t_lo), t_hi);
      int ti = (yc > ta1) + (yc > ta2) + (yc > ta3) + (yc > ta4);
      float2 ct = coef[8 + ti];
      float tau = fmaf(ct.y, yc, ct.x);

      // ring-buffer write (bank = tid%64 -> conflict-free for any idx)
      ring[ptr * BLOCK + tid] = x;

      // fractional delay read: rf = (ptr - D) mod L
      float D = fminf(fmaxf(dead * invDT, 0.0f), (float)(RING_L - 2));
      float rf = (float)ptr - D;
      rf += (rf < 0.0f) ? (float)RING_L : 0.0f;
      int i0 = (int)rf;            // rf >= 0 -> trunc == floor
      i0 = min(i0, RING_L - 1);    // guard rf rounding up to exactly L
      float a = rf - (float)i0;
      int i1 = i0 + 1;
      i1 = (i1 >= RING_L) ? 0 : i1;
      float v0 = ring[i0 * BLOCK + tid];
      float v1 = ring[i1 * BLOCK + tid];
      float delayed = fmaf(a, v1 - v0, v0);  // (1-a)*v0 + a*v1

      // first-order IIR; single v_rcp_f32 instead of full IEEE divide chain
      float af = DT_F * __builtin_amdgcn_rcpf(tau + DT_F);
      cur = fmaf(af, delayed - cur, cur);    // (1-af)*cur + af*delayed

      if (active) __builtin_nontemporal_store(cur, &out[t * N + g]);
      ptr = (ptr + 1 == RING_L) ? 0 : ptr + 1;
    }
  }
}

extern "C" void kernel_launch(void* const* d_in, const int* in_sizes, int n_in,
                              void* d_out, int out_size, void* d_ws, size_t ws_size,
                              hipStream_t stream) {
  const float* tp  = (const float*)d_in[0];  // target_pressure [T,E,C]
  const float* dax = (const float*)d_in[1];
  const float* dvl = (const float*)d_in[2];
  const float* tax = (const float*)d_in[3];
  const float* tvl = (const float*)d_in[4];
  float* out = (float*)d_out;

  const int T = 512;               // fixed by reference setup
  const int N = in_sizes[0] / T;   // E*C = 32768

  const int blocks = (N + BLOCK - 1) / BLOCK;
  const size_t shmem =
      (size_t)(2 * CHUNK * BLOCK + RING_L * BLOCK) * sizeof(float) +
      16 * sizeof(float2);

  pam_delay_kernel<<<blocks, BLOCK, shmem, stream>>>(tp, dax, dvl, tax, tvl,
                                                     out, T, N);
}